// DAGNNConv_54898271978224
// MI455X (gfx1250) — compile-verified
//
#include <hip/hip_runtime.h>
#include <hip/hip_bf16.h>

typedef __attribute__((ext_vector_type(2))) float v2f;
typedef __attribute__((ext_vector_type(8))) float v8f;
typedef __attribute__((ext_vector_type(4))) unsigned int u32x4;
typedef __attribute__((ext_vector_type(8))) int i32x8;
typedef __attribute__((ext_vector_type(4))) int i32x4;

#define F 128

// ---------------- zero fill ----------------
__global__ __launch_bounds__(256) void zero_f32(float* __restrict__ p, int n) {
    int i = blockIdx.x * 256 + threadIdx.x;
    if (i < n) p[i] = 0.0f;
}

__global__ __launch_bounds__(256) void zero_f32_v4(float4* __restrict__ p, long long n4) {
    long long i = (long long)blockIdx.x * 256 + threadIdx.x;
    if (i < n4) p[i] = make_float4(0.f, 0.f, 0.f, 0.f);
}

// ---------------- degree + norm ----------------
__global__ __launch_bounds__(256) void degree_kernel(const int* __restrict__ dst,
                                                     float* __restrict__ deg,
                                                     int E, int N) {
    int e = blockIdx.x * 256 + threadIdx.x;
    if (e < E) {
        int d = dst[e];
        if ((unsigned)d < (unsigned)N)
            unsafeAtomicAdd(deg + d, 1.0f);   // global_atomic_add_f32
    }
}

__global__ __launch_bounds__(256) void norm_kernel(float* __restrict__ nrm, int N) {
    int i = blockIdx.x * 256 + threadIdx.x;
    if (i < N) {
        float d = nrm[i];
        nrm[i] = (d > 0.0f) ? __frsqrt_rn(d) : 0.0f;
    }
}

// ---------------- SpMM hop: acc[dst] += norm[src] * h[src] ----------------
// one wave (32 lanes) per edge; lane handles a float4 => 128 floats per row.
// L2-resident gather + global_atomic_add_f32 scatter.
__global__ __launch_bounds__(256) void spmm_edge_kernel(const float* __restrict__ h,
                                                        const float* __restrict__ nrm,
                                                        float* __restrict__ acc,
                                                        const int* __restrict__ src,
                                                        const int* __restrict__ dst,
                                                        int E, int N) {
    int lane = threadIdx.x & 31;
    int e = blockIdx.x * 8 + (threadIdx.x >> 5);
    if (e >= E) return;
    int s = src[e];
    int d = dst[e];
    if ((unsigned)s >= (unsigned)N || (unsigned)d >= (unsigned)N) return;
    float ns = nrm[s];
    const float4* hv = (const float4*)(h + (size_t)s * F);
    float4 v = hv[lane];
    float* base = acc + (size_t)d * F + lane * 4;
    unsafeAtomicAdd(base + 0, v.x * ns);
    unsafeAtomicAdd(base + 1, v.y * ns);
    unsafeAtomicAdd(base + 2, v.z * ns);
    unsafeAtomicAdd(base + 3, v.w * ns);
}

// Issue one Tensor-DMA descriptor: stage a [rows x 128] f32 tile (row-major,
// stride 128) from global into LDS. Hardware zero-fills rows beyond
// tensor_dim1 (tail tiles). All fields wave-uniform (derived from blockIdx).
__device__ __forceinline__ void tdm_load_tile(const float* gsrc, unsigned lds_off,
                                              int rows_left) {
    unsigned long long ga = (unsigned long long)(uintptr_t)gsrc;
    u32x4 g0;
    g0.x = 1u;                                     // count=1, user mode, no gather
    g0.y = lds_off;                                // lds_addr (bytes)
    g0.z = (unsigned)ga;                           // global_addr[31:0]
    g0.w = ((unsigned)(ga >> 32) & 0x01FFFFFFu)    // global_addr[56:32]
           | (2u << 30);                           // type = 2 ("image")
    unsigned d1 = (unsigned)rows_left;             // tensor_dim1 (rows for OOB)
    i32x8 g1;
    g1[0] = (int)(2u << 16);                       // data_size = 2 -> 4 bytes
    g1[1] = (int)(128u << 16);                     // tensor_dim0[15:0] = 128
    g1[2] = (int)((d1 & 0xFFFFu) << 16);           // dim0[31:16]=0 | dim1[15:0]
    g1[3] = (int)(((d1 >> 16) & 0xFFFFu)           // dim1[31:16]
                  | (128u << 16));                 // tile_dim0 = 128
    g1[4] = 16;                                    // tile_dim1 = 16 rows
    g1[5] = 128;                                   // tensor_dim0_stride = 128
    g1[6] = 0;
    g1[7] = 0;
    i32x4 zz = {0, 0, 0, 0};
#if __has_include(<hip/amd_detail/amd_gfx1250_TDM.h>)
    i32x8 zz8 = {0, 0, 0, 0, 0, 0, 0, 0};
    __builtin_amdgcn_tensor_load_to_lds(g0, g1, zz, zz, zz8, 0);
#else
    __builtin_amdgcn_tensor_load_to_lds(g0, g1, zz, zz, 0);
#endif
}

// ---------------- combine: h = norm*acc (in place), out (+)= sigmoid(h.W)*h
// One wave per 16-node tile (1 wave/block so the TDM descriptor is uniform).
// Tile staged into LDS by the Tensor Data Mover; dot products via
// V_WMMA_F32_16X16X4_F32 with B[k][j] = W[4i+k] broadcast across columns, so
// every C column accumulates the per-node dot product over 32 K-chunks.
template <bool USE_NORM, bool WRITE_H, bool ACCUM>
__global__ __launch_bounds__(32) void combine_kernel(const float* __restrict__ hsrc,
                                                     float* __restrict__ hdst,
                                                     const float* __restrict__ nrm,
                                                     const float* __restrict__ W,
                                                     float* __restrict__ out,
                                                     int N) {
    __shared__ float tile[16 * F];   // 8 KB per workgroup (one wave)
    __shared__ float Wl[F];

    int lane = threadIdx.x;          // 0..31
    int n0 = blockIdx.x * 16;
    if (n0 >= N) return;

    // async tile DMA (TENSORcnt), overlap W staging with it
    tdm_load_tile(hsrc + (size_t)n0 * F,
                  (unsigned)(uintptr_t)&tile[0], N - n0);
#pragma unroll
    for (int j = 0; j < 4; ++j) Wl[lane * 4 + j] = W[lane * 4 + j];

    __builtin_amdgcn_s_wait_tensorcnt(0);
    __syncthreads();

    // A fragment (32-bit 16x4 layout): lanes 0-15 -> K=0,1 ; lanes 16-31 -> K=2,3
    int r = lane & 15;
    int arow = n0 + r;
    if (arow > N - 1) arow = N - 1;          // uniform-safe clamp, EXEC stays full
    float an = USE_NORM ? nrm[arow] : 1.0f;
    const float* abase = &tile[r * F];
    const int koff = (lane < 16) ? 0 : 2;    // A K-offset per lane half
    const int boff = (lane < 16) ? 0 : 1;    // B VGPR0 row per lane half (VGPR1=+2)

    v8f c = {0.f, 0.f, 0.f, 0.f, 0.f, 0.f, 0.f, 0.f};
#pragma unroll
    for (int i = 0; i < F / 4; ++i) {
        v2f a, b;
        a.x = abase[i * 4 + koff + 0] * an;
        a.y = abase[i * 4 + koff + 1] * an;
        b.x = Wl[i * 4 + boff + 0];
        b.y = Wl[i * 4 + boff + 2];
        c = __builtin_amdgcn_wmma_f32_16x16x4_f32(false, a, false, b,
                                                  (short)0, c, false, false);
    }

    // C layout: lanes 0-15 hold M=0..7 in VGPR0..7, lanes 16-31 hold M=8..15.
    // All columns equal -> lane 0 / lane 16 broadcast the 16 node dots.
    const float4* t4 = (const float4*)tile;
#pragma unroll
    for (int m = 0; m < 16; ++m) {
        float dot = __shfl(c[m & 7], (m < 8) ? 0 : 16, 32);
        float sgm = 1.0f / (1.0f + __expf(-dot));
        int row = n0 + m;
        if (row < N) {     // wave-uniform branch
            float rn = USE_NORM ? nrm[row] : 1.0f;
            float4 v = t4[m * 32 + lane];          // row from LDS, not global
            v.x *= rn; v.y *= rn; v.z *= rn; v.w *= rn;
            if (WRITE_H) {
                float4* hd = (float4*)(hdst + (size_t)row * F);
                hd[lane] = v;
            }
            float4* o4 = (float4*)(out + (size_t)row * F);
            float4 o;
            if (ACCUM) {
                o = o4[lane];
                o.x += sgm * v.x; o.y += sgm * v.y;
                o.z += sgm * v.z; o.w += sgm * v.w;
            } else {
                o.x = sgm * v.x; o.y = sgm * v.y;
                o.z = sgm * v.z; o.w = sgm * v.w;
            }
            o4[lane] = o;
        }
    }
}

extern "C" void kernel_launch(void* const* d_in, const int* in_sizes, int n_in,
                              void* d_out, int out_size, void* d_ws, size_t ws_size,
                              hipStream_t stream) {
    const float* x  = (const float*)d_in[0];
    const int* src  = (const int*)d_in[1];
    const int* dst  = (const int*)d_in[2];
    const float* W  = (const float*)d_in[3];
    float* out      = (float*)d_out;

    const int N = in_sizes[0] / F;
    const int E = in_sizes[1];
    const long long NF = (long long)N * F;

    // workspace layout: [ norm : N (padded) ][ buf0 : N*F ][ buf1 : N*F ]
    float* wsf  = (float*)d_ws;
    size_t napad = ((size_t)N + 63) & ~(size_t)63;
    float* nrm  = wsf;
    float* buf0 = wsf + napad;
    float* buf1 = buf0 + (size_t)N * F;

    // degrees -> norm = deg^-1/2
    zero_f32<<<(N + 255) / 256, 256, 0, stream>>>(nrm, N);
    degree_kernel<<<(E + 255) / 256, 256, 0, stream>>>(dst, nrm, E, N);
    norm_kernel<<<(N + 255) / 256, 256, 0, stream>>>(nrm, N);

    const int cgrid = (N + 15) / 16;             // 1 wave per 16-node tile
    const int egrid = (E + 7) / 8;               // 8 edges/block (1 wave each)
    const long long n4 = NF / 4;
    const int zgrid = (int)((n4 + 255) / 256);

    // k = 0 term: out = sigmoid(x.W) * x   (initializes d_out)
    combine_kernel<false, false, false><<<cgrid, 32, 0, stream>>>(
        x, nullptr, nullptr, W, out, N);

    // hops k = 1..10, ping-pong buffers
    for (int k = 1; k <= 10; ++k) {
        float* acc      = (k & 1) ? buf0 : buf1;
        const float* hs = (k == 1) ? x : ((k & 1) ? buf1 : buf0);
        zero_f32_v4<<<zgrid, 256, 0, stream>>>((float4*)acc, n4);
        spmm_edge_kernel<<<egrid, 256, 0, stream>>>(hs, nrm, acc, src, dst, E, N);
        combine_kernel<true, true, true><<<cgrid, 32, 0, stream>>>(
            acc, acc, nrm, W, out, N);
    }
}